// SelfAttention_68539088110358
// MI455X (gfx1250) — compile-verified
//
#include <hip/hip_runtime.h>
#include <math.h>

typedef float v2f __attribute__((ext_vector_type(2)));
typedef float v8f __attribute__((ext_vector_type(8)));

#define CHUNK 256

__device__ __forceinline__ v8f wmma_f32_k4(v2f a, v2f b, v8f c) {
  // D = A(16x4 f32) * B(4x16 f32) + C(16x16 f32)
  return __builtin_amdgcn_wmma_f32_16x16x4_f32(
      /*neg_a=*/false, a, /*neg_b=*/false, b,
      /*c_mod=*/(short)0, c, /*reuse_a=*/false, /*reuse_b=*/false);
}

// One 16-row block of y = W @ x per 128-thread block (4 waves split K, LDS reduce).
// A-tile layout (f32 16x4): lanes 0-15 hold K=k0+{0,1}, lanes 16-31 hold K=k0+{2,3}
// -> each lane does one b64 load at W[row]*dim + k0 + 2*(lane>>4).
// B is x broadcast to all 16 columns, so every column of D equals W@x; we
// extract column 0 from lane 0 (rows r0..r0+7) and lane 16 (rows r0+8..r0+15).
__device__ __forceinline__ void mv16_body(const float* __restrict__ W,
                                          const float* __restrict__ x,
                                          float* __restrict__ y,
                                          int dim, int r0) {
  const int lane = threadIdx.x & 31;
  const int wvid = threadIdx.x >> 5;   // 0..3: K-split segment
  const int half = lane >> 4;
  const int row  = r0 + (lane & 15);
  const int seg  = dim >> 2;
  const int kbeg = wvid * seg;
  const float* wr = W + (size_t)row * dim + 2 * half;
  const float* xp = x + 2 * half;

  v8f a0 = {}, a1 = {}, a2 = {}, a3 = {};
  for (int k0 = kbeg; k0 < kbeg + seg; k0 += 16) {
    v2f wa0 = *(const v2f*)(wr + k0);
    v2f wa1 = *(const v2f*)(wr + k0 + 4);
    v2f wa2 = *(const v2f*)(wr + k0 + 8);
    v2f wa3 = *(const v2f*)(wr + k0 + 12);
    v2f xb0 = *(const v2f*)(xp + k0);
    v2f xb1 = *(const v2f*)(xp + k0 + 4);
    v2f xb2 = *(const v2f*)(xp + k0 + 8);
    v2f xb3 = *(const v2f*)(xp + k0 + 12);
    a0 = wmma_f32_k4(wa0, xb0, a0);
    a1 = wmma_f32_k4(wa1, xb1, a1);
    a2 = wmma_f32_k4(wa2, xb2, a2);
    a3 = wmma_f32_k4(wa3, xb3, a3);
  }
  v8f acc = (a0 + a1) + (a2 + a3);

  __shared__ float red[4][16];
  if ((lane & 15) == 0) {
#pragma unroll
    for (int i = 0; i < 8; ++i) red[wvid][half * 8 + i] = acc[i];
  }
  __syncthreads();
  if (threadIdx.x < 16) {
    y[r0 + threadIdx.x] = (red[0][threadIdx.x] + red[1][threadIdx.x]) +
                          (red[2][threadIdx.x] + red[3][threadIdx.x]);
  }
}

// Phase 1: q = wq@x, k_new = wk@x, v_new = wv@x (grid.x = 3 * dim/16)
__global__ void __launch_bounds__(128)
qkv_kernel(const float* __restrict__ wq, const float* __restrict__ wk,
           const float* __restrict__ wv, const float* __restrict__ x,
           float* __restrict__ q, float* __restrict__ k, float* __restrict__ v,
           int dim) {
  const int rbTot = dim >> 4;
  const int m  = blockIdx.x / rbTot;
  const int rb = blockIdx.x % rbTot;
  const float* W = (m == 0) ? wq : (m == 1) ? wk : wv;
  float*       y = (m == 0) ? q  : (m == 1) ? k  : v;
  mv16_body(W, x, y, dim, rb * 16);
}

// Phase 4: final = wo @ attn_out (grid.x = dim/16)
__global__ void __launch_bounds__(128)
out_proj_kernel(const float* __restrict__ wo, const float* __restrict__ a,
                float* __restrict__ y, int dim) {
  mv16_body(wo, a, y, dim, blockIdx.x * 16);
}

// RoPE on q and k_new in-place (workspace only). One thread per (re,im) pair.
__global__ void rope_kernel(float* __restrict__ q, float* __restrict__ k,
                            const int* __restrict__ posp, int dim) {
  const int pid   = blockIdx.x * blockDim.x + threadIdx.x;
  const int npair = dim >> 1;
  if (pid >= 2 * npair) return;
  const int jp = pid % npair;
  float* v = (pid < npair ? q : k) + 2 * jp;
  const int j = ((2 * jp) & 127) >> 1;                 // pair index within head, D=128
  const float theta = expf(-(float)j * 0.14391156831212787f); // 10000^(-j/64)
  const float ang = (float)(*posp) * theta;
  const float c = cosf(ang), s = sinf(ang);
  const float v0 = v[0], v1 = v[1];
  v[0] = v0 * c - v1 * s;
  v[1] = v0 * s + v1 * c;
}

// Flash-decoding partial: block (head h, chunk c of CHUNK positions).
// Row p == pos is sourced from k_new/v_new (inputs are never mutated).
__global__ void __launch_bounds__(256)
attn_partial(const float* __restrict__ q, const float* __restrict__ knew,
             const float* __restrict__ vnew, const float* __restrict__ cache_k,
             const float* __restrict__ cache_v, const int* __restrict__ posp,
             float* __restrict__ cm, float* __restrict__ cs,
             float* __restrict__ co, int NC) {
  const int h = blockIdx.x, c = blockIdx.y;
  const int pos = *posp;
  const int pstart = c * CHUNK;
  if (pstart > pos) return;
  int pend = pstart + CHUNK;
  if (pend > pos + 1) pend = pos + 1;

  __shared__ float sq[128];
  __shared__ float se[CHUNK];
  __shared__ float red[8];
  __shared__ float bshare;

  const int tid = threadIdx.x, lane = tid & 31, wv = tid >> 5;
  if (tid < 128) sq[tid] = q[h * 128 + tid];
  se[tid] = -3.0e38f;
  __syncthreads();

  const float4 qv = ((const float4*)sq)[lane];
  for (int p = pstart + wv; p < pend; p += 8) {
    const float* kp = (p == pos) ? (knew + h * 128)
                                 : (cache_k + ((size_t)p * 32 + h) * 128);
    const float4 kv = ((const float4*)kp)[lane];
    float d = kv.x * qv.x + kv.y * qv.y + kv.z * qv.z + kv.w * qv.w;
#pragma unroll
    for (int off = 16; off > 0; off >>= 1) d += __shfl_xor(d, off, 32);
    if (lane == 0) se[p - pstart] = d * 0.08838834764831845f; // 1/sqrt(128)
  }
  __syncthreads();

  // block max
  float m = se[tid];
#pragma unroll
  for (int off = 16; off > 0; off >>= 1) m = fmaxf(m, __shfl_xor(m, off, 32));
  if (lane == 0) red[wv] = m;
  __syncthreads();
  if (tid == 0) {
    float M = red[0];
#pragma unroll
    for (int i = 1; i < 8; ++i) M = fmaxf(M, red[i]);
    bshare = M;
  }
  __syncthreads();
  const float M = bshare;

  const bool valid = (pstart + tid) < pend;
  const float e = valid ? expf(se[tid] - M) : 0.0f;
  se[tid] = e;
  // block sum of e
  float s = e;
#pragma unroll
  for (int off = 16; off > 0; off >>= 1) s += __shfl_xor(s, off, 32);
  if (lane == 0) red[wv] = s;
  __syncthreads();
  if (tid == 0) {
    float S = 0.0f;
#pragma unroll
    for (int i = 0; i < 8; ++i) S += red[i];
    cs[h * NC + c] = S;
    cm[h * NC + c] = M;
  }

  // partial weighted V sum (unnormalized): threads 0..127 own dimension d
  if (tid < 128) {
    float acc = 0.0f;
    for (int p = pstart; p < pend; ++p) {
      const float* vp = (p == pos) ? (vnew + h * 128)
                                   : (cache_v + ((size_t)p * 32 + h) * 128);
      acc += se[p - pstart] * vp[tid];
    }
    co[(h * NC + c) * 128 + tid] = acc;
  }
}

// Combine chunk partials with log-sum-exp rescale.
__global__ void __launch_bounds__(128)
attn_combine(const float* __restrict__ cm, const float* __restrict__ cs,
             const float* __restrict__ co, const int* __restrict__ posp,
             float* __restrict__ attn, int NC) {
  const int h = blockIdx.x, tid = threadIdx.x;
  const int pos = *posp;
  int nv = pos / CHUNK + 1;
  if (nv > NC) nv = NC;
  float M = -3.0e38f;
  for (int c = 0; c < nv; ++c) M = fmaxf(M, cm[h * NC + c]);
  float denom = 0.0f, acc = 0.0f;
  for (int c = 0; c < nv; ++c) {
    const float w = expf(cm[h * NC + c] - M);
    denom += w * cs[h * NC + c];
    acc   += w * co[(h * NC + c) * 128 + tid];
  }
  attn[h * 128 + tid] = acc / denom;
}

extern "C" void kernel_launch(void* const* d_in, const int* in_sizes, int n_in,
                              void* d_out, int out_size, void* d_ws, size_t ws_size,
                              hipStream_t stream) {
  (void)n_in; (void)out_size; (void)ws_size;
  const float* x    = (const float*)d_in[0];
  const float* wq   = (const float*)d_in[1];
  const float* wk   = (const float*)d_in[2];
  const float* wv   = (const float*)d_in[3];
  const float* wo   = (const float*)d_in[4];
  const float* ck   = (const float*)d_in[5];
  const float* cv   = (const float*)d_in[6];
  const int*   posp = (const int*)d_in[7];

  const int dim = in_sizes[0];              // 4096
  const int H = 32, D = 128;
  const int S  = in_sizes[5] / (H * D);     // 2048
  const int NC = (S + CHUNK - 1) / CHUNK;   // 8
  (void)D;

  float* ws  = (float*)d_ws;
  float* q   = ws;                // dim
  float* kn  = ws + dim;          // dim
  float* vn  = ws + 2 * dim;      // dim
  float* att = ws + 3 * dim;      // H*D == dim
  float* cm  = ws + 4 * dim;      // H*NC
  float* cs  = cm + H * NC;       // H*NC
  float* co  = cs + H * NC;       // H*NC*128

  const int rbTot = dim >> 4;
  qkv_kernel<<<3 * rbTot, 128, 0, stream>>>(wq, wk, wv, x, q, kn, vn, dim);
  rope_kernel<<<(dim + 255) / 256, 256, 0, stream>>>(q, kn, posp, dim);
  attn_partial<<<dim3(H, NC), 256, 0, stream>>>(q, kn, vn, ck, cv, posp, cm, cs, co, NC);
  attn_combine<<<H, 128, 0, stream>>>(cm, cs, co, posp, att, NC);
  out_proj_kernel<<<rbTot, 128, 0, stream>>>(wo, att, (float*)d_out, dim);
}